// CRD_6828998001543
// MI455X (gfx1250) — compile-verified
//
#include <hip/hip_runtime.h>

typedef float v2f __attribute__((ext_vector_type(2)));
typedef float v4f __attribute__((ext_vector_type(4)));
typedef float v8f __attribute__((ext_vector_type(8)));

// ---------------------------------------------------------------------------
// 1) deg init: self-loop contributes 1 to every node's degree
// ---------------------------------------------------------------------------
__global__ void deg_init_kernel(float* __restrict__ deg, int n) {
    int i = blockIdx.x * blockDim.x + threadIdx.x;
    if (i < n) deg[i] = 1.0f;
}

// ---------------------------------------------------------------------------
// 2) deg accumulation over edge destinations (hw fp32 atomic)
// ---------------------------------------------------------------------------
__global__ void deg_accum_kernel(const int* __restrict__ dst, float* __restrict__ deg,
                                 int n_edges) {
    int e = blockIdx.x * blockDim.x + threadIdx.x;
    if (e < n_edges) unsafeAtomicAdd(&deg[dst[e]], 1.0f);
}

// ---------------------------------------------------------------------------
// 3) dinv = deg > 0 ? rsqrt(deg) : 0   (in place)
// ---------------------------------------------------------------------------
__global__ void dinv_kernel(float* __restrict__ deg_dinv, int n) {
    int i = blockIdx.x * blockDim.x + threadIdx.x;
    if (i < n) {
        float d = deg_dinv[i];
        deg_dinv[i] = (d > 0.0f) ? rsqrtf(d) : 0.0f;
    }
}

// ---------------------------------------------------------------------------
// 4) h = x @ W via native fp32 WMMA (V_WMMA_F32_16X16X4_F32).
//    One wave (blockDim=32) computes a 16x32 output tile:
//      grid.x = row tiles (16 rows each), grid.y = d_out/32 column groups.
//    A-layout (16x4 fp32): lane L holds M = L&15, K-pair base (L>>4)*2
//      -> contiguous float2 load from x.
//    B-layout (4x16 fp32): lane L holds N = L&15, same K-pair base
//      -> two scalar loads from W (stride N between K rows; W is L2-hot).
//    C/D layout: VGPR r -> M = r + (L>=16 ? 8 : 0), N = L&15.
// ---------------------------------------------------------------------------
__global__ void gemm_wmma_f32_kernel(const float* __restrict__ x,
                                     const float* __restrict__ W,
                                     float* __restrict__ h,
                                     int n, int K, int N) {
    const int lane     = threadIdx.x;          // 0..31
    const int row_base = blockIdx.x * 16;
    const int col_base = blockIdx.y * 32;

    const int m  = lane & 15;
    const int kb = (lane >> 4) * 2;            // 0 or 2
    const int nc = lane & 15;                  // column within 16-wide tile

    int rowA = row_base + m;
    if (rowA >= n) rowA = n - 1;               // clamp; result discarded at store
    const float* xrow = x + (long)rowA * K;

    v8f c0 = {};
    v8f c1 = {};

    for (int k = 0; k < K; k += 4) {
        // A: two contiguous fp32 K-values for this lane's row
        v2f a = *(const v2f*)(xrow + k + kb);
        // B: same two K rows of W, this lane's column (two 16-wide tiles)
        const float* wr0 = W + (long)(k + kb) * N;
        const float* wr1 = wr0 + N;
        v2f b0, b1;
        b0.x = wr0[col_base + nc];
        b0.y = wr1[col_base + nc];
        b1.x = wr0[col_base + 16 + nc];
        b1.y = wr1[col_base + 16 + nc];

        c0 = __builtin_amdgcn_wmma_f32_16x16x4_f32(
                 false, a, false, b0, (short)0, c0, false, false);
        c1 = __builtin_amdgcn_wmma_f32_16x16x4_f32(
                 false, a, false, b1, (short)0, c1, false, false);
    }

    const int mOff = (lane >> 4) * 8;
#pragma unroll
    for (int r = 0; r < 8; ++r) {
        int row = row_base + mOff + r;
        if (row < n) {
            float* hr = h + (long)row * N + col_base + nc;
            hr[0]  = c0[r];
            hr[16] = c1[r];
        }
    }
}

// ---------------------------------------------------------------------------
// 5) out = h * dinv^2 (self-loop term) — also fully initializes d_out.
//    One block per node, blockDim = d_out (128).
// ---------------------------------------------------------------------------
__global__ void self_loop_kernel(const float* __restrict__ h,
                                 const float* __restrict__ dinv,
                                 float* __restrict__ out, int N) {
    int node = blockIdx.x;
    int f    = threadIdx.x;
    float di = dinv[node];
    long  i  = (long)node * N + f;
    out[i] = h[i] * di * di;
}

// ---------------------------------------------------------------------------
// 6) edge scatter: out[dst] += h[src] * dinv[src]*dinv[dst]
//    One wave per edge; each lane handles 4 contiguous floats (float4 gather,
//    4 hardware fp32 atomic adds). out (51 MB) lives in L2.
// ---------------------------------------------------------------------------
__global__ void edge_scatter_kernel(const int* __restrict__ src,
                                    const int* __restrict__ dst,
                                    const float* __restrict__ h,
                                    const float* __restrict__ dinv,
                                    float* __restrict__ out,
                                    int n_edges, int N) {
    int tid  = blockIdx.x * blockDim.x + threadIdx.x;
    int e    = tid >> 5;
    int lane = tid & 31;
    if (e >= n_edges) return;

    int   s    = src[e];
    int   d    = dst[e];
    float norm = dinv[s] * dinv[d];

    for (int f = lane * 4; f < N; f += 128) {
        v4f hv = *(const v4f*)(h + (long)s * N + f);
        float* op = out + (long)d * N + f;
        unsafeAtomicAdd(op + 0, hv.x * norm);
        unsafeAtomicAdd(op + 1, hv.y * norm);
        unsafeAtomicAdd(op + 2, hv.z * norm);
        unsafeAtomicAdd(op + 3, hv.w * norm);
    }
}

// ---------------------------------------------------------------------------
// 7) out = relu(out + b). One block per node, blockDim = d_out.
// ---------------------------------------------------------------------------
__global__ void bias_relu_kernel(float* __restrict__ out,
                                 const float* __restrict__ b, int N) {
    int node = blockIdx.x;
    int f    = threadIdx.x;
    long i   = (long)node * N + f;
    float v  = out[i] + b[f];
    out[i]   = v > 0.0f ? v : 0.0f;
}

// ---------------------------------------------------------------------------
extern "C" void kernel_launch(void* const* d_in, const int* in_sizes, int n_in,
                              void* d_out, int out_size, void* d_ws, size_t ws_size,
                              hipStream_t stream) {
    const float* x  = (const float*)d_in[0];
    const int*   ei = (const int*)d_in[1];     // [2, E]: row0 = src, row1 = dst
    const float* W  = (const float*)d_in[2];
    const float* b  = (const float*)d_in[3];
    float*       out = (float*)d_out;

    const int d_out_dim = in_sizes[3];                 // 128
    const int d_in_dim  = in_sizes[2] / d_out_dim;     // 256
    const int n_nodes   = in_sizes[0] / d_in_dim;      // 100000
    const int n_edges   = in_sizes[1] / 2;             // 1600000

    const int* src = ei;
    const int* dst = ei + n_edges;

    // workspace layout: h [n_nodes * d_out] | deg->dinv [n_nodes]
    float* h    = (float*)d_ws;
    float* dinv = h + (size_t)n_nodes * d_out_dim;

    // 1) deg = 1 (self-loops)
    {
        int blk = 256, grd = (n_nodes + blk - 1) / blk;
        deg_init_kernel<<<grd, blk, 0, stream>>>(dinv, n_nodes);
    }
    // 2) deg += scatter(ones, dst)
    {
        int blk = 256, grd = (n_edges + blk - 1) / blk;
        deg_accum_kernel<<<grd, blk, 0, stream>>>(dst, dinv, n_edges);
    }
    // 3) dinv = rsqrt(deg)
    {
        int blk = 256, grd = (n_nodes + blk - 1) / blk;
        dinv_kernel<<<grd, blk, 0, stream>>>(dinv, n_nodes);
    }
    // 4) h = x @ W (fp32 WMMA)
    {
        dim3 grd((n_nodes + 15) / 16, d_out_dim / 32);
        gemm_wmma_f32_kernel<<<grd, 32, 0, stream>>>(x, W, h, n_nodes, d_in_dim, d_out_dim);
    }
    // 5) out = h * dinv^2  (self-loop term; initializes all of d_out)
    self_loop_kernel<<<n_nodes, d_out_dim, 0, stream>>>(h, dinv, out, d_out_dim);
    // 6) out[dst] += h[src] * norm
    {
        long threads = (long)n_edges * 32;
        int  blk = 256;
        long grd = (threads + blk - 1) / blk;
        edge_scatter_kernel<<<(int)grd, blk, 0, stream>>>(src, dst, h, dinv, out,
                                                          n_edges, d_out_dim);
    }
    // 7) out = relu(out + b)
    bias_relu_kernel<<<n_nodes, d_out_dim, 0, stream>>>(out, b, d_out_dim);
}